// UnionLossWithCrossEntropyAndDiceAndBoundary_61555471286992
// MI455X (gfx1250) — compile-verified
//
#include <hip/hip_runtime.h>
#include <hip/hip_bf16.h>
#include <math.h>

// Problem constants (B,C,H,W) = (16,2,256,256), scalar f32 output.
#define BB   16
#define HH   256
#define WW   256
#define HW   (HH * WW)
#define NPIX (BB * HW)
#define BIGF 1.0e9f

typedef __attribute__((ext_vector_type(2))) float v2f;
typedef __attribute__((ext_vector_type(8))) float v8f;
typedef __attribute__((ext_vector_type(4))) int   v4i;

// gfx1250 async memory->LDS path (ASYNCcnt-tracked), guarded so a missing
// builtin falls back to synchronous loads instead of breaking the build.
#if defined(__has_builtin)
#if __has_builtin(__builtin_amdgcn_global_load_async_to_lds_b128) && \
    __has_builtin(__builtin_amdgcn_s_wait_asynccnt)
#define USE_ASYNC_LDS 1
#endif
#endif
#ifndef USE_ASYNC_LDS
#define USE_ASYNC_LDS 0
#endif

// Builtin signature (from hipcc diagnostic): param0 = addrspace(1) v4i*
// (global source), param1 = addrspace(3) v4i* (LDS destination), then two
// constant ints (offset, cpol).
#define GV4I(p) ((__attribute__((address_space(1))) v4i*)(p))
#define LV4I(p) ((__attribute__((address_space(3))) v4i*)(p))

// ---------------------------------------------------------------------------
// Wave-level sum using V_WMMA_F32_16X16X4_F32 (fp32 matrix path, exact).
// A = ones(16x4), B carries one value per lane (other slot 0).
// D[i][j] = sum_k B[k][j]  (every row of D equals the column sums),
// so c[0] holds the 16 column sums on lanes 0-15 (duplicated on 16-31).
// A 4-step xor-shuffle over 16 lanes yields the full 32-lane total.
// Requires EXEC == all ones: only call from fully-active 256-thread blocks.
// ---------------------------------------------------------------------------
__device__ __forceinline__ float wave_sum_wmma(float v) {
  v2f a; a.x = 1.0f; a.y = 1.0f;
  v2f b; b.x = v;    b.y = 0.0f;
  v8f c = {};
  c = __builtin_amdgcn_wmma_f32_16x16x4_f32(
      /*neg_a=*/false, a, /*neg_b=*/false, b,
      /*c_mod=*/(short)0, c, /*reuse_a=*/false, /*reuse_b=*/false);
  float s = c[0];
  s += __shfl_xor(s, 1, 32);
  s += __shfl_xor(s, 2, 32);
  s += __shfl_xor(s, 4, 32);
  s += __shfl_xor(s, 8, 32);
  return s;  // all lanes hold the wave total
}

// Block (256 thread) sum; result valid on threadIdx.x == 0.
__device__ __forceinline__ float block_sum(float v, float* sm /*8 floats*/) {
  float s = wave_sum_wmma(v);
  int lane = threadIdx.x & 31;
  int wv   = threadIdx.x >> 5;
  __syncthreads();              // protect sm across repeated calls
  if (lane == 0) sm[wv] = s;
  __syncthreads();
  float r = 0.f;
  if (threadIdx.x == 0) {
    for (int w = 0; w < 8; ++w) r += sm[w];
  }
  return r;
}

// ---------------------------------------------------------------------------
// Zero the scalar accumulators (ws[0..39]).
// ---------------------------------------------------------------------------
__global__ void k_zero(float* __restrict__ ws) {
  int i = threadIdx.x;
  if (i < 40) ws[i] = 0.f;
}

// ---------------------------------------------------------------------------
// Per-pixel: CE partial, dice sums (hard / t / hard*t), pos mask + pos count.
// grid = (HW/1024, B), block = 256, 4 pixels per thread.
// ws[0]=ce ws[1]=hard ws[2]=t ws[3]=hard*t ws[20+b]=pos count (per sample).
// ---------------------------------------------------------------------------
__global__ void k_pixel(const float* __restrict__ in, const int* __restrict__ tgt,
                        float* __restrict__ ws, float* __restrict__ pm) {
  __shared__ float sm[8];
  int b    = blockIdx.y;
  int base = blockIdx.x * 1024;
  const float* x0p = in + (size_t)b * 2 * HW;
  const float* x1p = x0p + HW;
  float ce = 0.f, hd = 0.f, tf = 0.f, ht = 0.f, ps = 0.f;
  for (int k = 0; k < 4; ++k) {
    int off  = base + k * 256 + threadIdx.x;
    if (k < 3) {  // prefetch next chunk (global_prefetch_b8)
      __builtin_prefetch(&x0p[off + 256], 0, 3);
      __builtin_prefetch(&x1p[off + 256], 0, 3);
    }
    float x0 = x0p[off];
    float x1 = x1p[off];
    int   t  = tgt[(size_t)b * HW + off];
    float m   = fmaxf(x0, x1);
    float lse = m + __logf(__expf(x0 - m) + __expf(x1 - m));
    float xt  = (t == 1) ? x1 : x0;
    ce += (lse - xt);
    float h  = (x1 >= x0) ? 1.f : 0.f;  // prob1 >= 0.5
    float p  = (x1 >  x0) ? 1.f : 0.f;  // argmax == 1 (ties -> class 0)
    float tv = (t == 1) ? 1.f : 0.f;
    hd += h; tf += tv; ht += h * tv; ps += p;
    pm[(size_t)b * HW + off] = p;
  }
  float r;
  r = block_sum(ce, sm); if (threadIdx.x == 0) atomicAdd(&ws[0], r);
  r = block_sum(hd, sm); if (threadIdx.x == 0) atomicAdd(&ws[1], r);
  r = block_sum(tf, sm); if (threadIdx.x == 0) atomicAdd(&ws[2], r);
  r = block_sum(ht, sm); if (threadIdx.x == 0) atomicAdd(&ws[3], r);
  r = block_sum(ps, sm); if (threadIdx.x == 0) atomicAdd(&ws[20 + b], r);
}

// ---------------------------------------------------------------------------
// Row-wise 1-D distance (fwd + bwd serial scans), squared result.
// g2n: row dist^2 to nearest pos pixel (EDT of neg), g2p: to nearest neg pixel.
// One thread per (b,row): 4096 threads.
// ---------------------------------------------------------------------------
__global__ void k_rowdist(const float* __restrict__ pm,
                          float* __restrict__ g2n, float* __restrict__ g2p) {
  int idx = blockIdx.x * blockDim.x + threadIdx.x;  // 0..B*H-1
  size_t base = (size_t)idx * WW;
  float cn = BIGF, cp = BIGF;
  for (int x = 0; x < WW; ++x) {
    float p = pm[base + x];
    cn = (p > 0.5f) ? 0.f : fminf(cn + 1.f, BIGF);  // neg mask zero at pos
    cp = (p > 0.5f) ? fminf(cp + 1.f, BIGF) : 0.f;  // pos mask zero at neg
    g2n[base + x] = cn;
    g2p[base + x] = cp;
  }
  cn = BIGF; cp = BIGF;
  for (int x = WW - 1; x >= 0; --x) {
    float p = pm[base + x];
    cn = (p > 0.5f) ? 0.f : fminf(cn + 1.f, BIGF);
    cp = (p > 0.5f) ? fminf(cp + 1.f, BIGF) : 0.f;
    float gn = fminf(cn, g2n[base + x]);
    float gp = fminf(cp, g2p[base + x]);
    g2n[base + x] = gn * gn;
    g2p[base + x] = gp * gp;
  }
}

// ---------------------------------------------------------------------------
// Column min-plus (exact EDT 2nd pass) + boundary dot-reduction.
// grid = (W/16, B); block = 256 (one thread per row i).
// LDS holds a 256x16 column tile of g2n/g2p, staged with
// GLOBAL_LOAD_ASYNC_TO_LDS_B128 (ASYNCcnt) when available.
// res = pos ? (1 - dp) : dn ; acc += t1 * res ; masked by has_pos.
// ---------------------------------------------------------------------------
__global__ void k_coldist(const float* __restrict__ g2n, const float* __restrict__ g2p,
                          const float* __restrict__ pm, const int* __restrict__ tgt,
                          float* __restrict__ ws) {
  __shared__ float ln[HH * 16];
  __shared__ float lp[HH * 16];
  __shared__ float sm[8];
  int b  = blockIdx.y;
  int j0 = blockIdx.x * 16;
  size_t sbase = (size_t)b * HW;
#if USE_ASYNC_LDS
  // 4096 floats per array = 1024 x B128; 4 async loads per thread per array.
  for (int q = threadIdx.x; q < (HH * 16) / 4; q += blockDim.x) {
    int r  = q >> 2;          // tile row
    int c4 = (q & 3) << 2;    // 4-float segment within the 16-wide row
    size_t g = sbase + (size_t)r * WW + j0 + c4;
    __builtin_amdgcn_global_load_async_to_lds_b128(GV4I(g2n + g), LV4I(ln + q * 4), 0, 0);
    __builtin_amdgcn_global_load_async_to_lds_b128(GV4I(g2p + g), LV4I(lp + q * 4), 0, 0);
  }
  __builtin_amdgcn_s_wait_asynccnt(0);
  __syncthreads();
#else
  for (int idx = threadIdx.x; idx < HH * 16; idx += blockDim.x) {
    int r = idx >> 4, c = idx & 15;
    size_t g = sbase + (size_t)r * WW + j0 + c;
    ln[idx] = g2n[g];
    lp[idx] = g2p[g];
  }
  __syncthreads();
#endif
  int   i  = threadIdx.x;
  float fi = (float)i;
  float haspos = (ws[20 + b] > 0.f) ? 1.f : 0.f;
  float acc = 0.f;
  for (int c = 0; c < 16; ++c) {
    float dn2 = 3.0e38f, dp2 = 3.0e38f;
    float dk = fi;
    for (int k = 0; k < HH; ++k) {
      float o = dk * dk;               // (i-k)^2
      dn2 = fminf(dn2, ln[k * 16 + c] + o);
      dp2 = fminf(dp2, lp[k * 16 + c] + o);
      dk -= 1.f;
    }
    float dn = sqrtf(dn2);
    float dp = sqrtf(dp2);
    size_t pix = sbase + (size_t)i * WW + j0 + c;
    float p  = pm[pix];
    float res = (p > 0.5f) ? (1.f - dp) : dn;
    float t1  = (tgt[pix] == 1) ? 1.f : 0.f;
    acc += t1 * res;
  }
  acc *= haspos;
  float r = block_sum(acc, sm);
  if (threadIdx.x == 0) atomicAdd(&ws[4 + b], r);
}

// ---------------------------------------------------------------------------
// Final scalar combine.
// ---------------------------------------------------------------------------
__global__ void k_final(const float* __restrict__ ws, float* __restrict__ out) {
  if (threadIdx.x == 0 && blockIdx.x == 0) {
    float ce   = ws[0] * (1.0f / (float)NPIX);
    float dice = 1.f - (2.f * ws[3] + 1.f) / (ws[1] + ws[2] + 1.f);
    float lb = 0.f;
    for (int b = 0; b < BB; ++b) lb += ws[4 + b];
    lb *= (1.0f / (float)HW);   // per-sample mean over H*W, summed over batch
    out[0] = ce + dice + lb * lb;
  }
}

// ---------------------------------------------------------------------------
// Launch: zero -> per-pixel -> row scans -> column min-plus -> combine.
// ws layout (floats): [0..39] accumulators, [64..) pos mask (N),
//                     then g2n (N), then g2p (N).  ~12.6 MB total.
// ---------------------------------------------------------------------------
extern "C" void kernel_launch(void* const* d_in, const int* in_sizes, int n_in,
                              void* d_out, int out_size, void* d_ws, size_t ws_size,
                              hipStream_t stream) {
  (void)in_sizes; (void)n_in; (void)out_size; (void)ws_size;
  const float* in  = (const float*)d_in[0];
  const int*   tgt = (const int*)d_in[1];
  // d_in[2] (starget), d_in[3] (dice_co), d_in[4] (boundary_co) are unused by
  // the reference's returned value.
  float* ws  = (float*)d_ws;
  float* pm  = ws + 64;
  float* g2n = pm + NPIX;
  float* g2p = g2n + NPIX;
  float* out = (float*)d_out;

  k_zero   <<<1, 64, 0, stream>>>(ws);
  k_pixel  <<<dim3(HW / 1024, BB), 256, 0, stream>>>(in, tgt, ws, pm);
  k_rowdist<<<dim3((BB * HH) / 256), 256, 0, stream>>>(pm, g2n, g2p);
  k_coldist<<<dim3(WW / 16, BB), 256, 0, stream>>>(g2n, g2p, pm, tgt, ws);
  k_final  <<<1, 32, 0, stream>>>(ws, out);
}